// SpikingNeuralNet_83202106458522
// MI455X (gfx1250) — compile-verified
//
#include <hip/hip_runtime.h>

#define NN    256   // neurons
#define NS    512   // samples
#define MS    32    // max spikes (rounds)
#define NSTEP 32    // RK4 steps per round
#define JN    8     // neurons per lane (wave32: 32*8 = 256)

typedef float v2f __attribute__((ext_vector_type(2)));
typedef float v8f __attribute__((ext_vector_type(8)));

__device__ __forceinline__ float sigf(float x) {
  // sigmoid(x) = 1 / (1 + exp(-x)); v_exp_f32 + v_rcp_f32
  return __builtin_amdgcn_rcpf(1.0f + __expf(-x));
}

__global__ __launch_bounds__(128) void snn_kernel(
    const float* __restrict__ ic_g, const float* __restrict__ w,
    const float* __restrict__ mu,   const float* __restrict__ v0,
    const float* __restrict__ i0,   const float* __restrict__ s0,
    const float* __restrict__ reset_s, const int* __restrict__ t1p,
    float* __restrict__ out)
{
  const int lane = threadIdx.x & 31;
  const int wid  = threadIdx.x >> 5;
  const int s    = blockIdx.x * 4 + wid;   // one wave == one sample

  const float mu1 = mu[0], mu2 = mu[1];
  const float t1f = (float)t1p[0];

  float* out_t = out;                                  // (NS, MS)
  float* out_v = out + (size_t)NS * MS;                // (NS, MS, NN, 3)
  float* out_m = out_v + (size_t)NS * MS * NN * 3;     // (NS, MS, NN)

  float ic[JN], vv[JN], ii[JN], ss[JN];
#pragma unroll
  for (int j = 0; j < JN; ++j) {
    const int n = lane + 32 * j;
    ic[j] = ic_g[n];
    vv[j] = v0[n];
    ii[j] = i0[n];
    ss[j] = s0[(size_t)s * NN + n];
  }

  float t0s = 0.0f;

  for (int r = 0; r < MS; ++r) {
    const float dt = (t1f - t0s) * (1.0f / NSTEP);
    const float h  = dt;
    float t   = t0s;
    float tev = t1f;
    bool  done = false;
    unsigned em = 0u;

    for (int st = 0; st < NSTEP; ++st) {
      float vn[JN], in2[JN], sn[JN];
#pragma unroll
      for (int j = 0; j < JN; ++j) {
        const float y0v = vv[j], y0i = ii[j], y0s = ss[j], icj = ic[j];
        const float k1v = mu1 * (y0i + icj - y0v);
        const float k1i = -mu2 * y0i;
        const float k1s = sigf(y0v);
        const float v2a = fmaf(0.5f * h, k1v, y0v);
        const float i2a = fmaf(0.5f * h, k1i, y0i);
        const float k2v = mu1 * (i2a + icj - v2a);
        const float k2i = -mu2 * i2a;
        const float k2s = sigf(v2a);
        const float v3a = fmaf(0.5f * h, k2v, y0v);
        const float i3a = fmaf(0.5f * h, k2i, y0i);
        const float k3v = mu1 * (i3a + icj - v3a);
        const float k3i = -mu2 * i3a;
        const float k3s = sigf(v3a);
        const float v4a = fmaf(h, k3v, y0v);
        const float i4a = fmaf(h, k3i, y0i);
        const float k4v = mu1 * (i4a + icj - v4a);
        const float k4i = -mu2 * i4a;
        const float k4s = sigf(v4a);
        const float h6 = h * (1.0f / 6.0f);
        vn[j]  = fmaf(h6, k1v + 2.0f * k2v + 2.0f * k3v + k4v, y0v);
        in2[j] = fmaf(h6, k1i + 2.0f * k2i + 2.0f * k3i + k4i, y0i);
        sn[j]  = fmaf(h6, k1s + 2.0f * k2s + 2.0f * k3s + k4s, y0s);
      }

      // event detection: any s_new > 0 ; argmax(s_new) with first-occurrence ties
      unsigned cr = 0u;
      float bsn = -3.402823466e38f, bsp = 0.0f;
      int   bidx = 0x7fffffff;
#pragma unroll
      for (int j = 0; j < JN; ++j) {
        cr |= (sn[j] > 0.0f ? 1u : 0u) << j;
        const bool g = sn[j] > bsn;   // strict > keeps smallest n within lane
        bsp  = g ? ss[j] : bsp;
        bidx = g ? (lane + 32 * j) : bidx;
        bsn  = g ? sn[j] : bsn;
      }
      const bool anyc = (__ballot(cr != 0u) != 0ull);   // wave-uniform

      if (anyc) {
#pragma unroll
        for (int off = 16; off >= 1; off >>= 1) {
          const float osn = __shfl_xor(bsn, off);
          const float osp = __shfl_xor(bsp, off);
          const int   oix = __shfl_xor(bidx, off);
          const bool take = (osn > bsn) || ((osn == bsn) && (oix < bidx));
          bsn  = take ? osn : bsn;
          bsp  = take ? osp : bsp;
          bidx = take ? oix : bidx;
        }
        float frac = bsp / (bsp - bsn + 1e-12f);
        frac = fminf(fmaxf(frac, 0.0f), 1.0f);
        tev = fmaf(frac, dt, t);
#pragma unroll
        for (int j = 0; j < JN; ++j) {
          vv[j] = fmaf(frac, vn[j]  - vv[j], vv[j]);
          ii[j] = fmaf(frac, in2[j] - ii[j], ii[j]);
          ss[j] = fmaf(frac, sn[j]  - ss[j], ss[j]);
        }
        em = cr;
        done = true;
        break;                       // remaining steps are no-ops in the reference
      } else {
#pragma unroll
        for (int j = 0; j < JN; ++j) { vv[j] = vn[j]; ii[j] = in2[j]; ss[j] = sn[j]; }
        t += dt;
      }
    }

    // ---- emit outputs for this round (yev == current y in both branches) ----
    const size_t ro = (size_t)s * MS + r;
    if (lane == 0) out_t[ro] = tev;
#pragma unroll
    for (int j = 0; j < JN; ++j) {
      const int n = lane + 32 * j;
      const size_t b = (ro * NN + n) * 3;
      out_v[b + 0] = vv[j];
      out_v[b + 1] = ii[j];
      out_v[b + 2] = ss[j];
      out_m[ro * NN + n] = ((em >> j) & 1u) ? 1.0f : 0.0f;
    }

    // ---- eidx = first neuron with emask set (wave min-reduction) ----
    int ei = 0x7fffffff;
    if (em) ei = lane + 32 * (__ffs(em) - 1);
#pragma unroll
    for (int off = 16; off >= 1; off >>= 1) {
      const int o = __shfl_xor(ei, off);
      ei = (o < ei) ? o : ei;
    }
    ei = __builtin_amdgcn_readfirstlane(ei);
    const bool dn = done;
    if (!dn) ei = 0;                 // keep B loads in-bounds; A=0 makes result 0
    const int kbase = ei & ~3;
    const int krem  = ei & 3;

    // ---- w[eidx] gather as exact one-hot WMMA: D = A(16x4) * W_tile(4x16) ----
    // A row 0 holds the one-hot at K = krem; rows 1..15 are zero.
    const float a0 = (dn && ((lane == 0 && krem == 0) || (lane == 16 && krem == 2))) ? 1.0f : 0.0f;
    const float a1 = (dn && ((lane == 0 && krem == 1) || (lane == 16 && krem == 3))) ? 1.0f : 0.0f;
    v2f A; A.x = a0; A.y = a1;

    const int bcol = lane & 15;
    const int rb   = kbase + ((lane >> 4) << 1);       // lanes 0-15: K+0/K+1, 16-31: K+2/K+3
    const float* wp = w + (size_t)rb * NN;

    float wr[JN];
#pragma unroll
    for (int j = 0; j < JN; ++j) {
      const int n0 = 32 * j;
      v2f Be; Be.x = wp[n0 + bcol];      Be.y = wp[NN + n0 + bcol];        // cols n0..n0+15
      v2f Bo; Bo.x = wp[n0 + 16 + bcol]; Bo.y = wp[NN + n0 + 16 + bcol];   // cols n0+16..n0+31
      v8f cz = {0.f, 0.f, 0.f, 0.f, 0.f, 0.f, 0.f, 0.f};
      v8f De = __builtin_amdgcn_wmma_f32_16x16x4_f32(false, A, false, Be, (short)0, cz, false, false);
      v8f Do = __builtin_amdgcn_wmma_f32_16x16x4_f32(false, A, false, Bo, (short)0, cz, false, false);
      // D row 0 (M=0) lives in VGPR0, lanes 0..15
      const float se = __shfl(De[0], bcol);
      const float so = __shfl(Do[0], bcol);
      wr[j] = (lane < 16) ? se : so;
    }

    // ---- next state ----
    const float* rs = reset_s + ((size_t)r * NS + s) * NN;
#pragma unroll
    for (int j = 0; j < JN; ++j) {
      const int n = lane + 32 * j;
      const bool e = (em >> j) & 1u;
      vv[j] = vv[j] - (e ? 1.0f : 0.0f);        // V_RESET = 1.0
      ii[j] = ii[j] + wr[j];                    // wr == 0 when !done (A was zero)
      ss[j] = fminf(e ? rs[n] : ss[j], 0.0f);
    }
    t0s = tev;
  }
}

extern "C" void kernel_launch(void* const* d_in, const int* in_sizes, int n_in,
                              void* d_out, int out_size, void* d_ws, size_t ws_size,
                              hipStream_t stream) {
  const float* ic = (const float*)d_in[0];
  const float* w  = (const float*)d_in[1];
  const float* mu = (const float*)d_in[2];
  const float* v0 = (const float*)d_in[3];
  const float* i0 = (const float*)d_in[4];
  const float* s0 = (const float*)d_in[5];
  const float* rs = (const float*)d_in[6];
  const int*   t1 = (const int*)d_in[7];
  // 128 blocks x 128 threads = 512 wave32 waves, one per sample
  snn_kernel<<<dim3(NS / 4), dim3(128), 0, stream>>>(ic, w, mu, v0, i0, s0, rs, t1,
                                                     (float*)d_out);
}